// LinearInterpolation_54254026883701
// MI455X (gfx1250) — compile-verified
//
#include <hip/hip_runtime.h>
#include <hip/hip_bf16.h>

typedef float v2f __attribute__((ext_vector_type(2)));
typedef float v8f __attribute__((ext_vector_type(8)));

#define BATCH_M 16384
#define KDIM    128
#define NDIM    4096

// One wave32 computes a 16(M) x 64(N) output strip with four 16x16 f32
// accumulators, using V_WMMA_F32_16X16X4_F32 over 32 K-slices (K=128).
// Block = 256 threads = 8 waves -> 16 x 512 per block.
// Grid = (16384/16, 4096/512) = (1024, 8). Exact tiling -> EXEC all ones.
__global__ __launch_bounds__(256) void
interp_gemm_wmma_f32(const float* __restrict__ x,
                     const float* __restrict__ w,
                     float* __restrict__ out) {
    const int wave = threadIdx.x >> 5;
    const int lane = threadIdx.x & 31;
    const int lrow = lane & 15;   // 0..15
    const int lhi  = lane >> 4;   // 0 or 1

    const int m0 = blockIdx.x * 16;                 // 0..16368
    const int n0 = blockIdx.y * 512 + wave * 64;    // 0..4032

    v8f acc0 = {}, acc1 = {}, acc2 = {}, acc3 = {};

    // A: lane holds x[m0+lrow][k0 + 2*lhi], x[m0+lrow][k0 + 2*lhi + 1]
    const float* xrow = x + (size_t)(m0 + lrow) * KDIM + 2 * lhi;
    // B: lane holds w[k0 + 2*lhi + {0,1}][n0 + lrow + 16*tile]
    const float* wbase = w + n0 + lrow;

    #pragma unroll 4
    for (int ks = 0; ks < KDIM / 4; ++ks) {
        const int k0 = ks * 4;

        v2f a = *(const v2f*)(xrow + k0);   // global_load_b64, 8B aligned

        const float* wp = wbase + (size_t)(k0 + 2 * lhi) * NDIM;
        v2f b0, b1, b2, b3;
        b0.x = wp[0];           b0.y = wp[NDIM];
        b1.x = wp[16];          b1.y = wp[NDIM + 16];
        b2.x = wp[32];          b2.y = wp[NDIM + 32];
        b3.x = wp[48];          b3.y = wp[NDIM + 48];

        acc0 = __builtin_amdgcn_wmma_f32_16x16x4_f32(
            false, a, false, b0, (short)0, acc0, false, false);
        acc1 = __builtin_amdgcn_wmma_f32_16x16x4_f32(
            false, a, false, b1, (short)0, acc1, false, false);
        acc2 = __builtin_amdgcn_wmma_f32_16x16x4_f32(
            false, a, false, b2, (short)0, acc2, false, false);
        acc3 = __builtin_amdgcn_wmma_f32_16x16x4_f32(
            false, a, false, b3, (short)0, acc3, false, false);
    }

    // C/D layout: vgpr v, lanes 0-15 -> row m0+v, lanes 16-31 -> row m0+v+8.
    // Output is 256 MB (larger than L2): use non-temporal stores so x/W stay
    // resident in the 192 MB L2 across the whole dispatch.
    float* orow = out + (size_t)(m0 + 8 * lhi) * NDIM + n0 + lrow;
    #pragma unroll
    for (int v = 0; v < 8; ++v) {
        float* p = orow + (size_t)v * NDIM;
        __builtin_nontemporal_store(acc0[v], p);
        __builtin_nontemporal_store(acc1[v], p + 16);
        __builtin_nontemporal_store(acc2[v], p + 32);
        __builtin_nontemporal_store(acc3[v], p + 48);
    }
}

extern "C" void kernel_launch(void* const* d_in, const int* in_sizes, int n_in,
                              void* d_out, int out_size, void* d_ws, size_t ws_size,
                              hipStream_t stream) {
    (void)in_sizes; (void)n_in; (void)d_ws; (void)ws_size; (void)out_size;
    const float* x = (const float*)d_in[0];          // (16384, 128) f32
    const float* w = (const float*)d_in[1];          // (128, 4096) f32
    float* out = (float*)d_out;                      // (16384, 4096) f32

    dim3 grid(BATCH_M / 16, NDIM / 512);             // (1024, 8)
    dim3 block(256);                                 // 8 waves of 32
    interp_gemm_wmma_f32<<<grid, block, 0, stream>>>(x, w, out);
}